// MutualAttention_81466939670699
// MI455X (gfx1250) — compile-verified
//
#include <hip/hip_runtime.h>

// MutualAttention, algebraically collapsed:
//   mean(softmax(S), axis=-1) == 1/N exactly, so
//   y[b] = W_eff @ h[(b-1)%B] + h[b],
//   W_eff = (W_out[:, :D] * (1/N) + W_out[:, D:] * (1 - 1/N)) @ W_g   (128x128)
// Main GEMM runs on V_WMMA_F32_16X16X4_F32 (full f32 precision).
// N/B/C/D are fixed by the reference -> hardcoded so every strided access
// lowers to base + 24-bit immediate offset (max row offset 127*16KB ~ 2MB < 8MB).

typedef __attribute__((ext_vector_type(2))) float v2f;
typedef __attribute__((ext_vector_type(8))) float v8f;

#define C_CH    128    // channels
#define D_DIM   64     // attention dim
#define B_BATCH 4      // batch
#define N_LEN   4096   // sequence length
#define TN      2      // 16-wide n-tiles per wave (A reused TN times)

// ---------------------------------------------------------------------------
// Kernel 1: fold the constant softmax-mean into the output projection and
// pre-multiply by W_g:  W_eff[ch, c] = sum_d M[ch, d] * W_g[d, c]
// with M[ch, d] = W_out[ch, d]*cf + W_out[ch, D+d]*cb.   (2 MFLOP, negligible)
// ---------------------------------------------------------------------------
__global__ void build_weff_kernel(const float* __restrict__ W_out,  // (C, 2D)
                                  const float* __restrict__ W_g,    // (D, C)
                                  float* __restrict__ W_eff,        // (C, C)
                                  float cf, float cb) {
    const int idx = blockIdx.x * blockDim.x + threadIdx.x;  // 0..16383
    const int ch = idx >> 7;
    const int c  = idx & 127;
    float acc = 0.0f;
    #pragma unroll 8
    for (int d = 0; d < D_DIM; ++d) {
        const float m = W_out[ch * (2 * D_DIM) + d] * cf +
                        W_out[ch * (2 * D_DIM) + D_DIM + d] * cb;
        acc = fmaf(m, W_g[d * C_CH + c], acc);
    }
    W_eff[ch * C_CH + c] = acc;
}

// ---------------------------------------------------------------------------
// Kernel 2: y[b] = W_eff @ h[(b-1)%B] + h[b], via V_WMMA_F32_16X16X4_F32.
// Block = 256 threads = 8 wave32; wave w owns channel tile [16w, 16w+16) for a
// 32-wide n strip (two 16x16 accumulators, A operand reused for both).
//
// VGPR layouts (ISA 7.12.2, wave32):
//   A 16x4 : lane L holds row m=L&15; VGPR0/1 = K = k0+2*(L>>4), +1
//   B 4x16 : lane L holds col n=L&15; VGPR0/1 = K = k0+2*(L>>4), +1
//   C 16x16: VGPR r, lane L -> row r + 8*(L>>4), col L&15
// Per-lane K/column components are folded into base pointers once; the fully
// unrolled K-loop then uses only compile-time immediate offsets.
// ---------------------------------------------------------------------------
__global__ void __launch_bounds__(256)
mutual_attn_gemm_kernel(const float* __restrict__ h,      // (B, C, N)
                        const float* __restrict__ W_eff,  // (C, C)
                        float* __restrict__ y) {          // (B, C, N)
    const int b     = blockIdx.y;
    const int bprev = (b + B_BATCH - 1) % B_BATCH;   // roll(h, 1, axis=0)
    const int n0    = blockIdx.x * (16 * TN);

    const int waveId = threadIdx.x >> 5;   // 0..7
    const int lane   = threadIdx.x & 31;
    const int half   = lane >> 4;          // 0: lanes 0-15, 1: lanes 16-31
    const int l16    = lane & 15;
    const int mBase  = waveId * 16;        // channel tile base

    // A base: row (mBase+l16), K offset 2*half folded in -> b64 pair at [k0]
    const float* __restrict__ Alane =
        W_eff + (size_t)(mBase + l16) * C_CH + 2 * half;
    // B base: batch bprev, column n0+l16, K offset 2*half rows folded in
    const float* __restrict__ Blane =
        h + (size_t)bprev * C_CH * N_LEN + (size_t)(2 * half) * N_LEN + n0 + l16;

    v8f acc0 = {}, acc1 = {};
    #pragma unroll
    for (int k0 = 0; k0 < C_CH; k0 += 4) {
        v2f av, b0, b1;
        av.x = Alane[k0];                       // contiguous -> global_load_b64
        av.y = Alane[k0 + 1];
        b0.x = Blane[k0 * N_LEN];               // all offsets: compile-time imm
        b0.y = Blane[k0 * N_LEN + N_LEN];
        b1.x = Blane[k0 * N_LEN + 16];
        b1.y = Blane[k0 * N_LEN + N_LEN + 16];
        acc0 = __builtin_amdgcn_wmma_f32_16x16x4_f32(
            false, av, false, b0, (short)0, acc0, false, false);
        acc1 = __builtin_amdgcn_wmma_f32_16x16x4_f32(
            false, av, false, b1, (short)0, acc1, false, false);
    }

    // Epilogue: row base folded once; r*N and +16 are immediate offsets.
    const size_t rowOff = (size_t)(mBase + 8 * half) * N_LEN + n0 + l16;
    const float* __restrict__ hres = h + (size_t)b * C_CH * N_LEN + rowOff;
    float* __restrict__ yrow       = y + (size_t)b * C_CH * N_LEN + rowOff;
    #pragma unroll
    for (int r = 0; r < 8; ++r) {
        yrow[r * N_LEN]      = acc0[r] + hres[r * N_LEN];
        yrow[r * N_LEN + 16] = acc1[r] + hres[r * N_LEN + 16];
    }
}

// ---------------------------------------------------------------------------
// Inputs (setup_inputs order): 0:h  1:W_theta  2:W_phi  3:W_g  4:W_out
// W_theta / W_phi are dead after the softmax-mean collapse.
// ---------------------------------------------------------------------------
extern "C" void kernel_launch(void* const* d_in, const int* in_sizes, int n_in,
                              void* d_out, int out_size, void* d_ws, size_t ws_size,
                              hipStream_t stream) {
    (void)in_sizes; (void)n_in; (void)out_size; (void)ws_size;
    const float* h     = (const float*)d_in[0];
    const float* W_g   = (const float*)d_in[3];
    const float* W_out = (const float*)d_in[4];
    float* y           = (float*)d_out;
    float* W_eff       = (float*)d_ws;          // 128*128*4 = 64 KB scratch

    const float cf = 1.0f / (float)N_LEN;       // exact mean of a softmax row
    const float cb = 1.0f - cf;

    build_weff_kernel<<<64, 256, 0, stream>>>(W_out, W_g, W_eff, cf, cb);

    dim3 grid(N_LEN / (16 * TN), B_BATCH);      // (128, 4) blocks x 8 waves
    mutual_attn_gemm_kernel<<<grid, 256, 0, stream>>>(h, W_eff, y);
}